// TopK_stablized_19756849562157
// MI455X (gfx1250) — compile-verified
//
#include <hip/hip_runtime.h>

typedef __attribute__((ext_vector_type(2))) float v2f;
typedef __attribute__((ext_vector_type(8))) float v8f;

#define N_SCORES 2048
#define BATCH    512
#define EPSR     0.1f
#define INV_EPS  10.0f
#define MAX_ITER 200
#define BLOCK    256
#define PT       8              // elements per thread (256*8 = 2048)
#define NWAVES   (BLOCK / 32)   // wave32 on gfx1250

// log(1/2048), log(512/2048), log(1536/2048)
#define LOG_MU  (-7.6246189861593985f)
#define LOG_NU0 (-1.3862943611198906f)
#define LOG_NU1 (-0.2876820724517809f)

__device__ __forceinline__ float softplus_f(float x) {
    // stable log(1 + e^x) = max(x,0) + log(1 + e^-|x|)
    float ax = __builtin_fabsf(x);
    return fmaxf(x, 0.0f) + __logf(1.0f + __expf(-ax));
}

// Sum a float across all 32 lanes using V_WMMA_F32_16X16X4_F32.
// A (16x4): lane m -> A[m,0]=a, A[m,1]=0 ; lane m+16 -> A[m,2]=a, A[m,3]=0.
// B = ones(4x16) => D[m,n] = a[m] + a[m+16] for every n.
// D VGPR j: lanes 0-15 hold row M=j, lanes 16-31 hold row M=j+8.
// Sum of 8 D VGPRs = half-sum; shfl_xor(16) folds halves -> total in all lanes.
__device__ __forceinline__ float wave_sum_wmma(float a) {
    v2f A; A.x = a;    A.y = 0.0f;
    v2f B; B.x = 1.0f; B.y = 1.0f;
    v8f C = {};
    v8f D = __builtin_amdgcn_wmma_f32_16x16x4_f32(false, A, false, B,
                                                  (short)0, C, false, false);
    float r = D[0] + D[1] + D[2] + D[3] + D[4] + D[5] + D[6] + D[7];
    r += __shfl_xor(r, 16, 32);
    return r;
}

__global__ void init_ws_kernel(unsigned* ws) {
    if (threadIdx.x == 0 && blockIdx.x == 0) ws[0] = 0u;
}

// Global Cmax = max over all (b,n) of max(s^2, (s-1)^2).  Values > 0, so
// float ordering == uint-bit ordering -> atomicMax on bits.
__global__ void __launch_bounds__(BLOCK)
cmax_kernel(const float* __restrict__ scores, unsigned* ws) {
    const int base = (blockIdx.x * BLOCK + threadIdx.x) * PT;
    const float4* p = (const float4*)(scores + base);
    float4 v0 = p[0], v1 = p[1];
    float sv[PT] = {v0.x, v0.y, v0.z, v0.w, v1.x, v1.y, v1.z, v1.w};
    float m = 0.0f;
#pragma unroll
    for (int i = 0; i < PT; ++i) {
        float s = sv[i];
        m = fmaxf(m, fmaxf(s * s, (s - 1.0f) * (s - 1.0f)));
    }
#pragma unroll
    for (int off = 16; off > 0; off >>= 1)
        m = fmaxf(m, __shfl_xor(m, off, 32));
    __shared__ float sm[NWAVES];
    if ((threadIdx.x & 31) == 0) sm[threadIdx.x >> 5] = m;
    __syncthreads();
    if (threadIdx.x == 0) {
        float mm = sm[0];
#pragma unroll
        for (int w = 1; w < NWAVES; ++w) mm = fmaxf(mm, sm[w]);
        atomicMax(ws, __float_as_uint(mm));
    }
}

__global__ void __launch_bounds__(BLOCK)
sinkhorn_kernel(const float* __restrict__ scores,
                const float* __restrict__ cmaxp,
                float* __restrict__ out) {
    const int b    = blockIdx.x;
    const int t    = threadIdx.x;
    const int lane = t & 31;
    const int wid  = t >> 5;

    const float cmax   = cmaxp[0];
    const float inv_ce = INV_EPS / cmax;   // 1/(cmax*eps)

    __shared__ float red0[NWAVES];
    __shared__ float red1[NWAVES];

    // ---- load scores, build d[n] = (C0-C1)/eps = (2s-1)/(cmax*eps) ----
    const float4* sp4 = (const float4*)(scores + (size_t)b * N_SCORES + t * PT);
    float4 v0 = sp4[0], v1 = sp4[1];
    float sv[PT] = {v0.x, v0.y, v0.z, v0.w, v1.x, v1.y, v1.z, v1.w};
    float d[PT];
    float dmn = 3.0e38f, dmx = -3.0e38f;
#pragma unroll
    for (int i = 0; i < PT; ++i) {
        d[i] = (2.0f * sv[i] - 1.0f) * inv_ce;
        dmn = fminf(dmn, d[i]);
        dmx = fmaxf(dmx, d[i]);
    }
#pragma unroll
    for (int off = 16; off > 0; off >>= 1) {
        dmn = fminf(dmn, __shfl_xor(dmn, off, 32));
        dmx = fmaxf(dmx, __shfl_xor(dmx, off, 32));
    }
    if (lane == 0) { red0[wid] = dmn; red1[wid] = dmx; }
    __syncthreads();
    float dmin = red0[0], dmax = red1[0];
#pragma unroll
    for (int w = 1; w < NWAVES; ++w) {
        dmin = fminf(dmin, red0[w]);
        dmax = fmaxf(dmax, red1[w]);
    }
    __syncthreads();

    // iteration-invariant: E[n] = exp(d[n] - dmax) <= 1
    float E[PT];
#pragma unroll
    for (int i = 0; i < PT; ++i) E[i] = __expf(d[i] - dmax);

    // ---- scalar Sinkhorn recursion on (g0, g1) ----
    float g0 = 1.0f, g1 = 1.0f;
    float g0p = 1.0f, dgp = 0.0f;   // values entering the last iteration

    for (int it = 0; it < MAX_ITER; ++it) {
        float dg = (g1 - g0) * INV_EPS;
        g0p = g0;
        dgp = dg;
        float spmin = softplus_f(dmin + dg);

        float a0 = 0.0f, a1 = 0.0f;
#pragma unroll
        for (int i = 0; i < PT; ++i) {
            float spx = softplus_f(d[i] + dg);
            float e0  = __expf(spmin - spx);   // in (0,1]
            a0 += e0;
            a1 += e0 * E[i];
        }

        // per-wave reduction via WMMA, then 8-wave fold through LDS
        float r0 = wave_sum_wmma(a0);
        float r1 = wave_sum_wmma(a1);
        if (lane == 0) { red0[wid] = r0; red1[wid] = r1; }
        __syncthreads();
        float S0 = 0.0f, SE = 0.0f;
#pragma unroll
        for (int w = 0; w < NWAVES; ++w) { S0 += red0[w]; SE += red1[w]; }
        __syncthreads();

        // g updates (f and the softplus(dmax) terms cancel analytically)
        float ng0 = g0 + EPSR * (LOG_NU0 - LOG_MU + spmin - __logf(S0));
        float ng1 = g0 + EPSR * (LOG_NU1 - LOG_MU - dmax + spmin - __logf(SE));
        g0 = ng0;
        g1 = ng1;
    }

    // ---- transport plan P = exp((-C + f + g)/eps), f from last iteration ----
    const float dgf = (g1 - g0) * INV_EPS;
    const float A0c = (g0 - g0p) * INV_EPS + LOG_MU;

    float p0[PT], p1[PT];
#pragma unroll
    for (int i = 0; i < PT; ++i) {
        float spx = softplus_f(d[i] + dgp);
        p0[i] = __expf(A0c - spx);                     // anchor 0
        p1[i] = __expf(A0c + d[i] + dgf - spx);        // anchor 1
    }

    float* o0 = out + (size_t)b * 2 * N_SCORES + t * PT;
    float* o1 = o0 + N_SCORES;
    ((float4*)o0)[0] = make_float4(p0[0], p0[1], p0[2], p0[3]);
    ((float4*)o0)[1] = make_float4(p0[4], p0[5], p0[6], p0[7]);
    ((float4*)o1)[0] = make_float4(p1[0], p1[1], p1[2], p1[3]);
    ((float4*)o1)[1] = make_float4(p1[4], p1[5], p1[6], p1[7]);
}

extern "C" void kernel_launch(void* const* d_in, const int* in_sizes, int n_in,
                              void* d_out, int out_size, void* d_ws, size_t ws_size,
                              hipStream_t stream) {
    (void)in_sizes; (void)n_in; (void)out_size; (void)ws_size;
    const float* scores = (const float*)d_in[0];
    float*       outp   = (float*)d_out;
    unsigned*    wsu    = (unsigned*)d_ws;

    init_ws_kernel<<<1, 1, 0, stream>>>(wsu);

    const int total_elems = BATCH * N_SCORES;
    const int cmax_blocks = total_elems / (BLOCK * PT);   // 512
    cmax_kernel<<<cmax_blocks, BLOCK, 0, stream>>>(scores, wsu);

    sinkhorn_kernel<<<BATCH, BLOCK, 0, stream>>>(scores, (const float*)d_ws, outp);
}